// ParameterFreeGCN_31791347925571
// MI455X (gfx1250) — compile-verified
//
#include <hip/hip_runtime.h>
#include <stdint.h>

// ParameterFreeGCN on MI455X (gfx1250, wave32).
// CSR build (deg -> scan -> fill) then node-per-wave gather kernels with
// register accumulation (no float atomics). Inter-layer rescale + residual +
// LayerNorm + next-layer pre-scale fused into the first gather kernel.
// Edge-index segments staged into LDS with the CDNA5 async-to-LDS engine
// (global_load_async_to_lds_b32, double-buffered, s_wait_asynccnt 1/0).
// Wave-uniform values forced to SGPRs via readfirstlane so row gathers use
// scalar-base addressing.

#define D_FEAT     128
#define LN_EPS     1e-5f
#define WPB        8      // waves per block in gather kernels (256 threads)
#define TPB        256
#define SCAN_TPB   256
#define SCAN_ELEMS 1024   // 4 elements per thread

typedef __attribute__((address_space(3))) int lds_i32_t;

__device__ __forceinline__ unsigned lds_byte_addr(int* p) {
  // generic -> LDS addrspace; ptrtoint gives the DS-visible byte offset
  return (unsigned)(uintptr_t)(lds_i32_t*)p;
}

__device__ __forceinline__ void async_idx_to_lds(unsigned lds_addr, const int* gptr) {
  asm volatile("global_load_async_to_lds_b32 %0, %1, off"
               :: "v"(lds_addr),
                  "v"((unsigned long long)(uintptr_t)gptr)
               : "memory");
}

__device__ __forceinline__ void wait_async_le0() {
  asm volatile("s_wait_asynccnt 0" ::: "memory");
}
__device__ __forceinline__ void wait_async_le1() {
  asm volatile("s_wait_asynccnt 1" ::: "memory");
}

// ---------------- degree / scan / CSR ----------------

__global__ void k_zero(unsigned* deg, int n) {
  int i = blockIdx.x * blockDim.x + threadIdx.x;
  if (i < n) deg[i] = 0u;
}

__global__ void k_degree(const int* __restrict__ dst, unsigned* deg, int e) {
  int i = blockIdx.x * blockDim.x + threadIdx.x;
  if (i < e) atomicAdd(&deg[dst[i]], 1u);
}

__global__ void k_blocksum(const unsigned* __restrict__ deg, unsigned* partial, int n) {
  __shared__ unsigned sm[SCAN_TPB];
  const int t = threadIdx.x;
  const int base = blockIdx.x * SCAN_ELEMS + t * 4;
  unsigned s = 0;
#pragma unroll
  for (int k = 0; k < 4; ++k) { int i = base + k; if (i < n) s += deg[i]; }
  sm[t] = s; __syncthreads();
  for (int off = SCAN_TPB / 2; off > 0; off >>= 1) {
    if (t < off) sm[t] += sm[t + off];
    __syncthreads();
  }
  if (t == 0) partial[blockIdx.x] = sm[0];
}

__global__ void k_scanpart(const unsigned* __restrict__ partial, unsigned* partOff, int nb) {
  __shared__ unsigned sm[512];
  const int t = threadIdx.x;
  sm[t] = (t < nb) ? partial[t] : 0u;
  __syncthreads();
  for (int off = 1; off < 512; off <<= 1) {
    unsigned x = (t >= off) ? sm[t - off] : 0u;
    __syncthreads();
    sm[t] += x;
    __syncthreads();
  }
  if (t < nb) partOff[t] = (t == 0) ? 0u : sm[t - 1];
}

__global__ void k_rowptr(const unsigned* __restrict__ deg, const unsigned* __restrict__ partOff,
                         unsigned* rowptr, unsigned* cursor, float* nrm, int n) {
  __shared__ unsigned sm[SCAN_TPB];
  const int t = threadIdx.x;
  const int base = blockIdx.x * SCAN_ELEMS + t * 4;
  unsigned v[4]; unsigned s = 0;
#pragma unroll
  for (int k = 0; k < 4; ++k) { int i = base + k; v[k] = (i < n) ? deg[i] : 0u; s += v[k]; }
  sm[t] = s; __syncthreads();
  for (int off = 1; off < SCAN_TPB; off <<= 1) {
    unsigned x = (t >= off) ? sm[t - off] : 0u;
    __syncthreads();
    sm[t] += x;
    __syncthreads();
  }
  unsigned excl = ((t == 0) ? 0u : sm[t - 1]) + partOff[blockIdx.x];
#pragma unroll
  for (int k = 0; k < 4; ++k) {
    int i = base + k;
    if (i < n) {
      rowptr[i] = excl;
      cursor[i] = excl;
      nrm[i]    = rsqrtf(1.0f + (float)v[k]);   // rsqrt(1 + deg)
      excl += v[k];
      if (i == n - 1) rowptr[n] = excl;         // == E
    }
  }
}

__global__ void k_fill(const int* __restrict__ dst, const int* __restrict__ src,
                       unsigned* cursor, int* colsrc, int e) {
  int i = blockIdx.x * blockDim.x + threadIdx.x;
  if (i < e) {
    int d = dst[i];
    unsigned slot = atomicAdd(&cursor[d], 1u);
    colsrc[slot] = src[i];
  }
}

// ---------------- node-wise kernels ----------------

// h1 = norm * x   (thread = one float4 of one row)
__global__ void k_h1(const float* __restrict__ x, const float* __restrict__ nrm,
                     float* __restrict__ h, int n) {
  int tid = blockIdx.x * blockDim.x + threadIdx.x;   // n*32 float4's
  if (tid >= n * 32) return;
  int node = tid >> 5;
  float nv = nrm[node];
  float4 v = reinterpret_cast<const float4*>(x)[tid];
  v.x *= nv; v.y *= nv; v.z *= nv; v.w *= nv;
  reinterpret_cast<float4*>(h)[tid] = v;
}

// Gather + accumulate a node's in-edge rows. Indices are staged into LDS with
// double-buffered async-to-LDS loads; consumed wave-uniformly (readfirstlane)
// so the 512B row loads use scalar-base addressing.
__device__ __forceinline__ float4 gather_acc(const float* __restrict__ h,
                                             unsigned beg, unsigned end,
                                             const int* __restrict__ colsrc,
                                             int lane, int* slot0, int* slot1,
                                             float4 acc)
{
  if (beg >= end) return acc;
  const unsigned a0 = lds_byte_addr(slot0 + lane);
  const unsigned a1 = lds_byte_addr(slot1 + lane);
  {
    unsigned i  = beg + (unsigned)lane;
    unsigned ic = (i < end) ? i : (end - 1u);    // clamp: all 32 lanes issue
    async_idx_to_lds(a0, colsrc + ic);
  }
  int buf = 0;
  for (unsigned base = beg; base < end; base += 32u) {
    const unsigned next = base + 32u;
    if (next < end) {                            // prefetch next chunk
      unsigned i  = next + (unsigned)lane;
      unsigned ic = (i < end) ? i : (end - 1u);
      async_idx_to_lds(buf ? a0 : a1, colsrc + ic);
      wait_async_le1();                          // older chunk done (in-order)
    } else {
      wait_async_le0();
    }
    int* cur = buf ? slot1 : slot0;
    const int cnt = (int)(((end - base) < 32u) ? (end - base) : 32u);
    for (int k = 0; k < cnt; ++k) {
      int s = __builtin_amdgcn_readfirstlane(cur[k]);   // wave-uniform index
      float4 v = reinterpret_cast<const float4*>(h + (size_t)s * D_FEAT)[lane];
      acc.x += v.x; acc.y += v.y; acc.z += v.z; acc.w += v.w;
    }
    buf ^= 1;
  }
  return acc;
}

// Fused: conv1 aggregate -> y1 = norm*agg -> t = y1 + ori -> LayerNorm(t)
//        -> h2 = norm * LN(t)   (layer-2 pre-scale)
__global__ void __launch_bounds__(TPB)
k_conv1mid(const float* __restrict__ h1, const float* __restrict__ x0,
           const float* __restrict__ nrm, const unsigned* __restrict__ rowptr,
           const int* __restrict__ colsrc, float* __restrict__ h2, int n)
{
  __shared__ int idxbuf[WPB][2][32];
  const int w = threadIdx.x >> 5, lane = threadIdx.x & 31;
  const int nid = __builtin_amdgcn_readfirstlane(blockIdx.x * WPB + w);
  if (nid >= n) return;

  float4 acc = reinterpret_cast<const float4*>(h1 + (size_t)nid * D_FEAT)[lane];
  const unsigned beg = rowptr[nid], end = rowptr[nid + 1];
  acc = gather_acc(h1, beg, end, colsrc, lane, &idxbuf[w][0][0], &idxbuf[w][1][0], acc);

  const float nv = nrm[nid];
  const float4 o = reinterpret_cast<const float4*>(x0 + (size_t)nid * D_FEAT)[lane];
  float4 t;
  t.x = nv * acc.x + o.x;  t.y = nv * acc.y + o.y;
  t.z = nv * acc.z + o.z;  t.w = nv * acc.w + o.w;

  float s1 = t.x + t.y + t.z + t.w;
  float s2 = t.x * t.x + t.y * t.y + t.z * t.z + t.w * t.w;
  for (int off = 16; off > 0; off >>= 1) {             // wave32 reduction
    s1 += __shfl_xor(s1, off, 32);
    s2 += __shfl_xor(s2, off, 32);
  }
  const float mu  = s1 * (1.0f / (float)D_FEAT);
  const float var = s2 * (1.0f / (float)D_FEAT) - mu * mu;
  const float r   = rsqrtf(var + LN_EPS);

  float4 hn;
  hn.x = nv * ((t.x - mu) * r);  hn.y = nv * ((t.y - mu) * r);
  hn.z = nv * ((t.z - mu) * r);  hn.w = nv * ((t.w - mu) * r);
  reinterpret_cast<float4*>(h2 + (size_t)nid * D_FEAT)[lane] = hn;
}

// Fused: conv2 aggregate -> out = norm*agg + ori
__global__ void __launch_bounds__(TPB)
k_conv2fin(const float* __restrict__ h2, const float* __restrict__ x0,
           const float* __restrict__ nrm, const unsigned* __restrict__ rowptr,
           const int* __restrict__ colsrc, float* __restrict__ out, int n)
{
  __shared__ int idxbuf[WPB][2][32];
  const int w = threadIdx.x >> 5, lane = threadIdx.x & 31;
  const int nid = __builtin_amdgcn_readfirstlane(blockIdx.x * WPB + w);
  if (nid >= n) return;

  float4 acc = reinterpret_cast<const float4*>(h2 + (size_t)nid * D_FEAT)[lane];
  const unsigned beg = rowptr[nid], end = rowptr[nid + 1];
  acc = gather_acc(h2, beg, end, colsrc, lane, &idxbuf[w][0][0], &idxbuf[w][1][0], acc);

  const float nv = nrm[nid];
  const float4 o = reinterpret_cast<const float4*>(x0 + (size_t)nid * D_FEAT)[lane];
  float4 r;
  r.x = nv * acc.x + o.x;  r.y = nv * acc.y + o.y;
  r.z = nv * acc.z + o.z;  r.w = nv * acc.w + o.w;
  reinterpret_cast<float4*>(out + (size_t)nid * D_FEAT)[lane] = r;
}

// ---------------- launch ----------------

extern "C" void kernel_launch(void* const* d_in, const int* in_sizes, int n_in,
                              void* d_out, int out_size, void* d_ws, size_t ws_size,
                              hipStream_t stream) {
  const float* x  = (const float*)d_in[0];
  const int*   ei = (const int*)d_in[1];        // int32 (JAX default x64-off)
  const int N = in_sizes[0] / D_FEAT;           // 100000
  const int E = in_sizes[1] / 2;                // 1600000
  const int* dst = ei;                          // edge_index[0]
  const int* src = ei + E;                      // edge_index[1]
  float* out = (float*)d_out;

  // workspace carve (256B aligned), total ~110.6 MB
  char* p = (char*)d_ws;
  auto carve = [&](size_t bytes) -> void* {
    void* r = (void*)p;
    p += (bytes + 255) & ~(size_t)255;
    return r;
  };
  unsigned* deg     = (unsigned*)carve((size_t)N * 4);
  unsigned* cursor  = (unsigned*)carve((size_t)N * 4);
  unsigned* rowptr  = (unsigned*)carve((size_t)(N + 1) * 4);
  float*    nrm     = (float*)   carve((size_t)N * 4);
  unsigned* partial = (unsigned*)carve(512 * 4);
  unsigned* partOff = (unsigned*)carve(512 * 4);
  int*      colsrc  = (int*)     carve((size_t)E * 4);
  float*    h1      = (float*)   carve((size_t)N * D_FEAT * 4);
  float*    h2      = (float*)   carve((size_t)N * D_FEAT * 4);

  const int nb = (N + SCAN_ELEMS - 1) / SCAN_ELEMS;      // 98 (<512)

  k_zero    <<<(N + 255) / 256, 256, 0, stream>>>(deg, N);
  k_degree  <<<(E + 255) / 256, 256, 0, stream>>>(dst, deg, E);
  k_blocksum<<<nb, SCAN_TPB, 0, stream>>>(deg, partial, N);
  k_scanpart<<<1, 512, 0, stream>>>(partial, partOff, nb);
  k_rowptr  <<<nb, SCAN_TPB, 0, stream>>>(deg, partOff, rowptr, cursor, nrm, N);
  k_fill    <<<(E + 255) / 256, 256, 0, stream>>>(dst, src, cursor, colsrc, E);
  k_h1      <<<(N * 32 + 255) / 256, 256, 0, stream>>>(x, nrm, h1, N);

  const int nblk = (N + WPB - 1) / WPB;                  // 12500 blocks x 8 waves
  k_conv1mid<<<nblk, TPB, 0, stream>>>(h1, x, nrm, rowptr, colsrc, h2, N);
  k_conv2fin<<<nblk, TPB, 0, stream>>>(h2, x, nrm, rowptr, colsrc, out, N);
}